// Contextual_Attention_Module_76579266888401
// MI455X (gfx1250) — compile-verified
//
#include <hip/hip_runtime.h>
#include <math.h>

typedef float v2f __attribute__((ext_vector_type(2)));
typedef float v8f __attribute__((ext_vector_type(8)));

#define BATCH   4
#define CCH     64
#define HF      96
#define PF      98      // padded full-res (halo 1)
#define HDS     48
#define PD      50      // padded downsampled (halo 1)
#define LTOT    2304    // 48*48
#define PROWS   2500    // 50*50 padded attention rows
#define CST_DS  2500    // channel stride, padded ds image
#define CST_F   9604    // channel stride, padded full image
#define SCALE_F 10.0f

// ---------------------------------------------------------------------------
// WMMA wrapper: D = A(16x4,f32) * B(4x16,f32) + C(16x16,f32)
// ---------------------------------------------------------------------------
__device__ __forceinline__ v8f wmma_f32(v2f a, v2f b, v8f c) {
  return __builtin_amdgcn_wmma_f32_16x16x4_f32(false, a, false, b, (short)0, c,
                                               false, false);
}

// ---------------------------------------------------------------------------
// K1: stride-2 downsample of f and b into zero-padded [B][64][50][50]
// ---------------------------------------------------------------------------
__global__ void pad_downsample_kernel(const float* __restrict__ f,
                                      const float* __restrict__ b,
                                      float* __restrict__ fdp,
                                      float* __restrict__ bdp) {
  int idx = blockIdx.x * blockDim.x + threadIdx.x;
  const int total = BATCH * CCH * PD * PD;
  if (idx >= total) return;
  int x = idx % PD;
  int t = idx / PD;
  int y = t % PD;  t /= PD;
  int c = t % CCH;
  int bb = t / CCH;
  float fv = 0.0f, bv = 0.0f;
  if (y >= 1 && y <= HDS && x >= 1 && x <= HDS) {
    size_t src = ((size_t)(bb * CCH + c) * HF + 2 * (y - 1)) * HF + 2 * (x - 1);
    fv = f[src];
    bv = b[src];
  }
  fdp[idx] = fv;
  bdp[idx] = bv;
}

// ---------------------------------------------------------------------------
// K2: zero-padded copy of full-res b into [B][64][98][98]
// ---------------------------------------------------------------------------
__global__ void pad_full_kernel(const float* __restrict__ b,
                                float* __restrict__ bp) {
  int idx = blockIdx.x * blockDim.x + threadIdx.x;
  const int total = BATCH * CCH * PF * PF;
  if (idx >= total) return;
  int x = idx % PF;
  int t = idx / PF;
  int y = t % PF;  t /= PF;
  int c = t % CCH;
  int bb = t / CCH;
  float v = 0.0f;
  if (y >= 1 && y <= HF && x >= 1 && x <= HF)
    v = b[((size_t)(bb * CCH + c) * HF + (y - 1)) * HF + (x - 1)];
  bp[idx] = v;
}

// ---------------------------------------------------------------------------
// K3: per-patch inverse L2 norm of bd patches (padded input, no checks) + mask
// ---------------------------------------------------------------------------
__global__ void norm_mask_kernel(const float* __restrict__ bdp,
                                 const float* __restrict__ mask,
                                 float* __restrict__ ninv,
                                 float* __restrict__ mmv) {
  int idx = blockIdx.x * blockDim.x + threadIdx.x;
  if (idx >= BATCH * LTOT) return;
  int bb = idx / LTOT, l = idx - bb * LTOT;
  int ly = l / HDS, lx = l - ly * HDS;
  const float* base = bdp + (size_t)bb * CCH * CST_DS + ly * PD + lx;
  float s = 0.0f;
  for (int c = 0; c < CCH; ++c) {
    const float* pc = base + c * CST_DS;
    for (int u = 0; u < 3; ++u)
      for (int v = 0; v < 3; ++v) {
        float w = pc[u * PD + v];
        s += w * w;
      }
  }
  ninv[idx] = 1.0f / fmaxf(sqrtf(s), 1e-4f);

  const float* mk = mask + (size_t)bb * HDS * HDS;
  float msum = 0.0f;
  for (int u = 0; u < 3; ++u) {
    int yy = ly + u - 1;
    if ((unsigned)yy >= (unsigned)HDS) continue;
    for (int v = 0; v < 3; ++v) {
      int xx = lx + v - 1;
      if ((unsigned)xx >= (unsigned)HDS) continue;
      msum += mk[yy * HDS + xx];
    }
  }
  mmv[idx] = (msum == 0.0f) ? 1.0f : 0.0f;
}

// ---------------------------------------------------------------------------
// K4: correlation GEMM. s_raw[b][p][l] = <patch_fd(p), patch_bd(l)> * ninv[l]
// One wave = 16(p) x 64(l) tile, M=p, N=l, K=9 taps x 64 ch.
// Hot loop: unconditional loads with linear channel stride + 4 WMMAs.
// ---------------------------------------------------------------------------
__global__ __launch_bounds__(32) void corr_kernel(const float* __restrict__ fdp,
                                                  const float* __restrict__ bdp,
                                                  const float* __restrict__ ninv,
                                                  float* __restrict__ sraw) {
  int lane = threadIdx.x;
  int m = lane & 15, hi = lane >> 4;
  int bb = blockIdx.z;
  int p_base = blockIdx.y * 16;
  int l_base = blockIdx.x * 64;

  int p = p_base + m;
  int py = p / HDS, px = p - py * HDS;

  int boff[4];
#pragma unroll
  for (int j = 0; j < 4; ++j) {
    int l = l_base + j * 16 + m;
    int ly = l / HDS, lx = l - ly * HDS;
    boff[j] = ly * PD + lx;
  }

  const float* fb = fdp + (size_t)bb * CCH * CST_DS;
  const float* bbase = bdp + (size_t)bb * CCH * CST_DS;

  v8f acc[4] = {};
  for (int tap = 0; tap < 9; ++tap) {
    int u = tap / 3, v = tap - u * 3;
    const float* aP = fb + py * PD + px + u * PD + v + 2 * hi * CST_DS;
    const float* bP0 = bbase + boff[0] + u * PD + v + 2 * hi * CST_DS;
    const float* bP1 = bbase + boff[1] + u * PD + v + 2 * hi * CST_DS;
    const float* bP2 = bbase + boff[2] + u * PD + v + 2 * hi * CST_DS;
    const float* bP3 = bbase + boff[3] + u * PD + v + 2 * hi * CST_DS;
#pragma unroll 4
    for (int cb = 0; cb < CCH; cb += 4) {
      int o0 = cb * CST_DS, o1 = o0 + CST_DS;
      v2f a  = { aP[o0],  aP[o1]  };
      v2f b0 = { bP0[o0], bP0[o1] };
      v2f b1 = { bP1[o0], bP1[o1] };
      v2f b2 = { bP2[o0], bP2[o1] };
      v2f b3 = { bP3[o0], bP3[o1] };
      acc[0] = wmma_f32(a, b0, acc[0]);
      acc[1] = wmma_f32(a, b1, acc[1]);
      acc[2] = wmma_f32(a, b2, acc[2]);
      acc[3] = wmma_f32(a, b3, acc[3]);
    }
  }

#pragma unroll
  for (int j = 0; j < 4; ++j) {
    int n = l_base + j * 16 + m;
    float sc = ninv[bb * LTOT + n];
#pragma unroll
    for (int r = 0; r < 8; ++r) {
      int row = p_base + hi * 8 + r;
      sraw[((size_t)bb * LTOT + row) * LTOT + n] = acc[j][r] * sc;
    }
  }
}

// ---------------------------------------------------------------------------
// K5a: zero the halo rows of the padded attention buffer [B][2500][2304]
// ---------------------------------------------------------------------------
__global__ __launch_bounds__(256) void sfp_halo_zero_kernel(float* __restrict__ sfp) {
  int r  = blockIdx.x;     // padded row 0..2499
  int bb = blockIdx.y;
  int iy = r / PD - 1, ix = r % PD - 1;
  if ((unsigned)iy < (unsigned)HDS && (unsigned)ix < (unsigned)HDS) return;
  float* row = sfp + ((size_t)bb * PROWS + r) * LTOT;
  for (int l = threadIdx.x; l < LTOT; l += 256) row[l] = 0.0f;
}

// ---------------------------------------------------------------------------
// K5b: both fuse passes collapsed to one 9-tap gather + pre-softmax mask mult.
// Writes padded rows: row((y,x)) = (y+1)*50 + (x+1)
// ---------------------------------------------------------------------------
__global__ void fuse_mask_kernel(const float* __restrict__ sraw,
                                 const float* __restrict__ mmv,
                                 float* __restrict__ sfp) {
  int l = blockIdx.x * blockDim.x + threadIdx.x;
  int p = blockIdx.y;
  int bb = blockIdx.z;
  if (l >= LTOT) return;
  int y = p / HDS, x = p - HDS * y;
  int ly = l / HDS, lx = l - HDS * ly;
  int q  = x * HDS + y;
  int mi = lx * HDS + ly;
  const float* sb = sraw + (size_t)bb * LTOT * LTOT;
  float sum = 0.0f;
  for (int d2 = -1; d2 <= 1; ++d2) {
    int q2 = q + d2, m2 = mi + d2;
    if ((unsigned)q2 >= (unsigned)LTOT || (unsigned)m2 >= (unsigned)LTOT) continue;
    int x2 = q2 / HDS, y2 = q2 - HDS * x2;
    int lx2 = m2 / HDS, ly2 = m2 - HDS * lx2;
    int p2 = y2 * HDS + x2;
    int l2 = ly2 * HDS + lx2;
    for (int d1 = -1; d1 <= 1; ++d1) {
      int p3 = p2 + d1, l3 = l2 + d1;
      if ((unsigned)p3 >= (unsigned)LTOT || (unsigned)l3 >= (unsigned)LTOT) continue;
      sum += sb[(size_t)p3 * LTOT + l3];
    }
  }
  int r = (y + 1) * PD + (x + 1);
  sfp[((size_t)bb * PROWS + r) * LTOT + l] = sum * mmv[bb * LTOT + l];
}

// ---------------------------------------------------------------------------
// K6: per-pixel softmax over 2304 patches (in place, padded rows) + argmax
// ---------------------------------------------------------------------------
__global__ __launch_bounds__(256) void softmax_argmax_kernel(float* __restrict__ sfp,
                                                             const float* __restrict__ mmv,
                                                             float* __restrict__ off_out) {
  __shared__ float red[256];
  __shared__ float redv[256];
  __shared__ int   redi[256];
  int tid = threadIdx.x;
  int p   = blockIdx.x;
  int bb  = blockIdx.y;
  int y = p / HDS, x = p - HDS * y;
  int r = (y + 1) * PD + (x + 1);
  float* row = sfp + ((size_t)bb * PROWS + r) * LTOT;
  const float* mrow = mmv + bb * LTOT;

  float vmax = -INFINITY;
  for (int l = tid; l < LTOT; l += 256) vmax = fmaxf(vmax, row[l]);
  red[tid] = vmax;
  __syncthreads();
  for (int st = 128; st > 0; st >>= 1) {
    if (tid < st) red[tid] = fmaxf(red[tid], red[tid + st]);
    __syncthreads();
  }
  vmax = red[0];
  __syncthreads();

  float sum = 0.0f;
  float bestv = -INFINITY;
  int   besti = 0x7fffffff;
  for (int l = tid; l < LTOT; l += 256) {
    float v = row[l];
    sum += __expf(SCALE_F * (v - vmax));
    if (mrow[l] != 0.0f) {
      if (v > bestv || (v == bestv && l < besti)) { bestv = v; besti = l; }
    }
  }
  red[tid]  = sum;
  redv[tid] = bestv;
  redi[tid] = besti;
  __syncthreads();
  for (int st = 128; st > 0; st >>= 1) {
    if (tid < st) {
      red[tid] += red[tid + st];
      float v2 = redv[tid + st];
      int   i2 = redi[tid + st];
      if (v2 > redv[tid] || (v2 == redv[tid] && i2 < redi[tid])) {
        redv[tid] = v2;
        redi[tid] = i2;
      }
    }
    __syncthreads();
  }
  float inv = 1.0f / red[0];
  int l_star = (redv[0] == -INFINITY) ? 0 : redi[0];

  for (int l = tid; l < LTOT; l += 256)
    row[l] = __expf(SCALE_F * (row[l] - vmax)) * inv * mrow[l];

  if (tid == 0) {
    int lys = l_star / HDS, lxs = l_star - HDS * lys;
    off_out[((size_t)(bb * 2) + 0) * LTOT + p] = (float)(lys - y);
    off_out[((size_t)(bb * 2) + 1) * LTOT + p] = (float)(lxs - x);
  }
}

// ---------------------------------------------------------------------------
// K7: zero halo of a padded [B][64][98][98] image
// ---------------------------------------------------------------------------
__global__ void img_halo_zero_kernel(float* __restrict__ buf) {
  int idx = blockIdx.x * blockDim.x + threadIdx.x;
  const int total = BATCH * CCH * PF * PF;
  if (idx >= total) return;
  int x = idx % PF;
  int y = (idx / PF) % PF;
  if (y == 0 || y == PF - 1 || x == 0 || x == PF - 1) buf[idx] = 0.0f;
}

// ---------------------------------------------------------------------------
// K8: stride-2 transposed conv via WMMA. 4 parity classes; per class
// M=2304 output pixels, N=64 ch, K=2304 patches x 4 taps.
// A: unconditional aligned float2 loads from padded attention rows.
// B: linear-in-lx gather from padded b image.
// ---------------------------------------------------------------------------
__global__ __launch_bounds__(32) void deconv_kernel(const float* __restrict__ sfp,
                                                    const float* __restrict__ bp,
                                                    float* __restrict__ y0p) {
  int lane = threadIdx.x;
  int m = lane & 15, hi = lane >> 4;
  int bb  = blockIdx.z;
  int cls = blockIdx.y;
  int ry = cls >> 1, rx = cls & 1;
  int tile = blockIdx.x;

  int ab = tile * 16 + m;
  int a  = ab / HDS, bc = ab - a * HDS;

  const float* sb = sfp + (size_t)bb * PROWS * LTOT;
  const float* bib = bp + (size_t)bb * CCH * CST_F;

  int co[4];
#pragma unroll
  for (int jf = 0; jf < 4; ++jf) co[jf] = jf * 16 + m;

  v8f acc[4] = {};
  for (int i = 0; i < 2; ++i) {
    int iy = a + ry - 1 + i;          // -1..48, always valid in padded rows
    int ku = 3 - ry - 2 * i;
    for (int j = 0; j < 2; ++j) {
      int ix = bc + rx - 1 + j;
      int kv = 3 - rx - 2 * j;
      const float* srow = sb + (size_t)((iy + 1) * PD + (ix + 1)) * LTOT + 2 * hi;
      __builtin_prefetch(srow, 0, 1);
      for (int ly = 0; ly < HDS; ++ly) {
        const float* arow = srow + ly * HDS;
        const float* br0 = bib + (co[0] * PF + 2 * ly + ku) * PF + kv + 4 * hi;
        const float* br1 = bib + (co[1] * PF + 2 * ly + ku) * PF + kv + 4 * hi;
        const float* br2 = bib + (co[2] * PF + 2 * ly + ku) * PF + kv + 4 * hi;
        const float* br3 = bib + (co[3] * PF + 2 * ly + ku) * PF + kv + 4 * hi;
#pragma unroll
        for (int lxb = 0; lxb < HDS; lxb += 4) {
          v2f av = *(const v2f*)(arow + lxb);
          v2f b0 = { br0[2 * lxb], br0[2 * lxb + 2] };
          v2f b1 = { br1[2 * lxb], br1[2 * lxb + 2] };
          v2f b2 = { br2[2 * lxb], br2[2 * lxb + 2] };
          v2f b3 = { br3[2 * lxb], br3[2 * lxb + 2] };
          acc[0] = wmma_f32(av, b0, acc[0]);
          acc[1] = wmma_f32(av, b1, acc[1]);
          acc[2] = wmma_f32(av, b2, acc[2]);
          acc[3] = wmma_f32(av, b3, acc[3]);
        }
      }
    }
  }

#pragma unroll
  for (int jf = 0; jf < 4; ++jf) {
#pragma unroll
    for (int r = 0; r < 8; ++r) {
      int ab2 = tile * 16 + hi * 8 + r;
      int a2 = ab2 / HDS, b2 = ab2 - a2 * HDS;
      int oy = 2 * a2 + ry, ox = 2 * b2 + rx;
      y0p[((size_t)(bb * CCH + co[jf]) * PF + oy + 1) * PF + ox + 1] =
          acc[jf][r] * 0.25f;
    }
  }
}

// ---------------------------------------------------------------------------
// K9: 3x3 conv 64->64 + bias + ELU via WMMA, padded input.
// out stride/halo parametrized: (98,1) -> padded ws buffer, (96,0) -> d_out.
// ---------------------------------------------------------------------------
__global__ __launch_bounds__(32) void conv3_elu_kernel(const float* __restrict__ inp,
                                                       const float* __restrict__ w,
                                                       const float* __restrict__ bias,
                                                       float* __restrict__ out,
                                                       int os, int oh) {
  int lane = threadIdx.x;
  int m = lane & 15, hi = lane >> 4;
  int bb = blockIdx.z;
  int tile = blockIdx.x;
  int p = tile * 16 + m;
  int y = p / HF, x = p - HF * y;

  const float* ib = inp + (size_t)bb * CCH * CST_F + 2 * hi * CST_F;

  int n[4];
#pragma unroll
  for (int jf = 0; jf < 4; ++jf) n[jf] = jf * 16 + m;

  v8f acc[4] = {};
  for (int tap = 0; tap < 9; ++tap) {
    int u = tap / 3, v = tap - u * 3;
    const float* aP = ib + (y + u) * PF + (x + v);
    const float* w0 = w + (n[0] * CCH + 2 * hi) * 9 + tap;
    const float* w1 = w + (n[1] * CCH + 2 * hi) * 9 + tap;
    const float* w2 = w + (n[2] * CCH + 2 * hi) * 9 + tap;
    const float* w3 = w + (n[3] * CCH + 2 * hi) * 9 + tap;
#pragma unroll 4
    for (int cb = 0; cb < CCH; cb += 4) {
      int a0 = cb * CST_F, a1 = a0 + CST_F;
      int c0 = cb * 9,     c1 = c0 + 9;
      v2f av = { aP[a0], aP[a1] };
      v2f b0 = { w0[c0], w0[c1] };
      v2f b1 = { w1[c0], w1[c1] };
      v2f b2 = { w2[c0], w2[c1] };
      v2f b3 = { w3[c0], w3[c1] };
      acc[0] = wmma_f32(av, b0, acc[0]);
      acc[1] = wmma_f32(av, b1, acc[1]);
      acc[2] = wmma_f32(av, b2, acc[2]);
      acc[3] = wmma_f32(av, b3, acc[3]);
    }
  }

#pragma unroll
  for (int jf = 0; jf < 4; ++jf) {
    float bsv = bias[n[jf]];
#pragma unroll
    for (int r = 0; r < 8; ++r) {
      int p2 = tile * 16 + hi * 8 + r;
      int y2 = p2 / HF, x2 = p2 - HF * y2;
      float v = acc[jf][r] + bsv;
      v = (v > 0.0f) ? v : (__expf(v) - 1.0f);   // ELU
      out[((size_t)(bb * CCH + n[jf]) * os + y2 + oh) * os + x2 + oh] = v;
    }
  }
}

// ---------------------------------------------------------------------------
extern "C" void kernel_launch(void* const* d_in, const int* in_sizes, int n_in,
                              void* d_out, int out_size, void* d_ws, size_t ws_size,
                              hipStream_t stream) {
  (void)in_sizes; (void)n_in; (void)out_size; (void)ws_size;

  const float* f    = (const float*)d_in[0];
  const float* b    = (const float*)d_in[1];
  const float* mask = (const float*)d_in[2];
  const float* w1   = (const float*)d_in[3];
  const float* b1   = (const float*)d_in[4];
  const float* w2   = (const float*)d_in[5];
  const float* b2   = (const float*)d_in[6];

  float* out = (float*)d_out;
  float* ws  = (float*)d_ws;

  const size_t n_dsp = (size_t)BATCH * CCH * PD * PD;    // 640000
  const size_t n_fp  = (size_t)BATCH * CCH * PF * PF;    // 2458624
  const size_t n_bl  = (size_t)BATCH * LTOT;             // 9216
  const size_t n_s   = (size_t)BATCH * LTOT * LTOT;      // 21233664
  const size_t n_sp  = (size_t)BATCH * PROWS * LTOT;     // 23040000

  float* fdp  = ws;
  float* bdp  = fdp + n_dsp;
  float* bpd  = bdp + n_dsp;
  float* ninv = bpd + n_fp;
  float* mmv  = ninv + n_bl;
  float* sraw = mmv + n_bl;
  float* sfp  = sraw + n_s;
  float* y0p  = sfp + n_sp;
  float* y1p  = y0p + n_fp;

  float* off_out = out + (size_t)BATCH * CCH * HF * HF;

  pad_downsample_kernel<<<(int)((n_dsp + 255) / 256), 256, 0, stream>>>(f, b, fdp, bdp);
  pad_full_kernel<<<(int)((n_fp + 255) / 256), 256, 0, stream>>>(b, bpd);
  norm_mask_kernel<<<(int)((n_bl + 255) / 256), 256, 0, stream>>>(bdp, mask, ninv, mmv);
  corr_kernel<<<dim3(LTOT / 64, LTOT / 16, BATCH), 32, 0, stream>>>(fdp, bdp, ninv, sraw);
  sfp_halo_zero_kernel<<<dim3(PROWS, BATCH), 256, 0, stream>>>(sfp);
  fuse_mask_kernel<<<dim3(LTOT / 256, LTOT, BATCH), 256, 0, stream>>>(sraw, mmv, sfp);
  softmax_argmax_kernel<<<dim3(LTOT, BATCH), 256, 0, stream>>>(sfp, mmv, off_out);
  img_halo_zero_kernel<<<(int)((n_fp + 255) / 256), 256, 0, stream>>>(y0p);
  img_halo_zero_kernel<<<(int)((n_fp + 255) / 256), 256, 0, stream>>>(y1p);
  deconv_kernel<<<dim3(LTOT / 16, 4, BATCH), 32, 0, stream>>>(sfp, bpd, y0p);
  conv3_elu_kernel<<<dim3((HF * HF) / 16, 1, BATCH), 32, 0, stream>>>(y0p, w1, b1, y1p, PF, 1);
  conv3_elu_kernel<<<dim3((HF * HF) / 16, 1, BATCH), 32, 0, stream>>>(y1p, w2, b2, out, HF, 0);
}